// SpatialRegressor3_81853486727431
// MI455X (gfx1250) — compile-verified
//
#include <hip/hip_runtime.h>

typedef __attribute__((ext_vector_type(2))) float v2f;
typedef __attribute__((ext_vector_type(8))) float v8f;

#define EPSBN 1e-5f
#define LDSW 36   // LDS tile row stride (floats): conflict-free for 64 banks

// ---------------- ws layout (float offsets) ----------------
enum {
  OFF_SU  = 0,     // u stats: sum[4], sq[4]
  OFF_SO1 = 8,     // o1 stats: sum[32], sq[32]
  OFF_SA1 = 72,
  OFF_SR  = 136,   // (o1+o2) stats
  OFF_SA2 = 200,
  OFF_SX  = 264,   // pooled x stats
  OFF_ST1 = 328,
  OFF_ST2 = 392,
  OFF_ST3 = 456,
  OFF_CP1 = 520,   // coeffs: s[C], t[C]
  OFF_CO1 = 528,
  OFF_CP2 = 536,
  OFF_CO2 = 600,
  OFF_CP3 = 664,
  OFF_CO3 = 728,
  OFF_CT1 = 792,
  OFF_CT2 = 856,
  OFF_CT3 = 920,
  OFF_CT4 = 984,
  OFF_STATS_END = 1048,
  OFF_X  = 1048,            // 4096*32
  OFF_T1 = OFF_X  + 131072,
  OFF_T2 = OFF_T1 + 131072,
  OFF_T3 = OFF_T2 + 131072,
};

__device__ __forceinline__ v8f v8zero() {
  v8f z;
  #pragma unroll
  for (int i = 0; i < 8; i++) z[i] = 0.f;
  return z;
}

// Branch-free tanh: 1 - 2/(exp(2x)+1); saturates to +-1, uses v_exp/v_rcp
__device__ __forceinline__ float fast_tanh(float x) {
  float e = __expf(2.0f * x);
  return 1.0f - 2.0f * __builtin_amdgcn_rcpf(e + 1.0f);
}

__device__ __forceinline__ v8f wmma4(v2f a, v2f b, v8f c) {
  // D = A(16x4) * B(4x16) + C   (f32, wave32)
  return __builtin_amdgcn_wmma_f32_16x16x4_f32(false, a, false, b, (short)0, c,
                                               false, false);
}

// BN apply to a K-pair: coef layout s[C], t[C]; col2 = column of first elem
__device__ __forceinline__ v2f bn2v(v2f raw, const float* coef, int C, int col2) {
  v2f s = *(const v2f*)(coef + col2);
  v2f t = *(const v2f*)(coef + C + col2);
  v2f r; r.x = raw.x * s.x + t.x; r.y = raw.y * s.y + t.y;
  return r;
}

// Layer with K=4 straight from u: one WMMA per 16x16 tile
__device__ __forceinline__ void layer1(const float* __restrict__ u,
                                       const float* __restrict__ w1,
                                       const float* coef, int r0, int lane,
                                       v8f d[2]) {
  int r = lane & 15;
  int koff = (lane >> 4) * 2;
  v2f raw = *(const v2f*)(u + (r0 + r) * 4 + koff);
  v2f a = bn2v(raw, coef, 4, koff);
  v2f b0 = *(const v2f*)(w1 + r * 4 + koff);
  v2f b1 = *(const v2f*)(w1 + (16 + r) * 4 + koff);
  v8f z = v8zero();
  d[0] = wmma4(a, b0, z);
  d[1] = wmma4(a, b1, z);
  #pragma unroll
  for (int t = 0; t < 2; t++)
    #pragma unroll
    for (int i = 0; i < 8; i++) d[t][i] = fast_tanh(d[t][i]);
}

// K=32 GEMM: 8 chained WMMAs per column tile; input tile in LDS (stride LDSW)
__device__ __forceinline__ void gemm32(const float* xl, const float* wl,
                                       const float* coef, int lane, v8f d[2]) {
  int r = lane & 15;
  int koff = (lane >> 4) * 2;
  d[0] = v8zero(); d[1] = v8zero();
  #pragma unroll
  for (int k = 0; k < 8; k++) {
    int cc = 4 * k + koff;
    v2f raw = *(const v2f*)(xl + r * LDSW + cc);
    v2f a = bn2v(raw, coef, 32, cc);
    v2f b0 = *(const v2f*)(wl + r * 32 + cc);
    v2f b1 = *(const v2f*)(wl + (16 + r) * 32 + cc);
    d[0] = wmma4(a, b0, d[0]);
    d[1] = wmma4(a, b1, d[1]);
  }
}

// Scatter D-fragment (lane owns fixed column, 8 rows) into LDS tile
__device__ __forceinline__ void store_tile(float* xl, int lane, int n0, v8f d) {
  int col = n0 + (lane & 15);
  int rb = (lane >> 4) * 8;
  #pragma unroll
  for (int j = 0; j < 8; j++) xl[(rb + j) * LDSW + col] = d[j];
}

// Per-column sum / sum-of-squares from D pair into shared st[64]
__device__ __forceinline__ void stats_acc(float* st, int lane, const v8f d[2]) {
  #pragma unroll
  for (int t = 0; t < 2; t++) {
    int col = t * 16 + (lane & 15);
    float s = 0.f, q = 0.f;
    #pragma unroll
    for (int j = 0; j < 8; j++) { float v = d[t][j]; s += v; q += v * v; }
    atomicAdd(&st[col], s);
    atomicAdd(&st[32 + col], q);
  }
}

// ---------------- kernels ----------------
__global__ void k_zero(float* ws) {
  for (int i = threadIdx.x; i < OFF_STATS_END; i += blockDim.x) ws[i] = 0.f;
}

__global__ void k_stats_u(const float* __restrict__ u, float* __restrict__ ws,
                          int total) {
  __shared__ float s[8];
  if (threadIdx.x < 8) s[threadIdx.x] = 0.f;
  __syncthreads();
  int gid = blockIdx.x * blockDim.x + threadIdx.x;
  int stride = gridDim.x * blockDim.x;   // multiple of 4 -> feature fixed
  int f = gid & 3;
  float sum = 0.f, sq = 0.f;
  for (int i = gid; i < total; i += stride) { float v = u[i]; sum += v; sq += v * v; }
  atomicAdd(&s[f], sum);
  atomicAdd(&s[4 + f], sq);
  __syncthreads();
  if (threadIdx.x < 8) atomicAdd(&ws[OFF_SU + threadIdx.x], s[threadIdx.x]);
}

__global__ void k_coef(float* ws, int statsOff, const float* __restrict__ g,
                       const float* __restrict__ b, int coefOff, int C, float inv) {
  int c = threadIdx.x;
  if (c < C) {
    float mean = ws[statsOff + c] * inv;
    float var = ws[statsOff + C + c] * inv - mean * mean;
    float s = g[c] * rsqrtf(var + EPSBN);
    ws[coefOff + c] = s;
    ws[coefOff + C + c] = b[c] - mean * s;
  }
}

__global__ void __launch_bounds__(256) k_pass2(const float* __restrict__ u,
    const float* __restrict__ pw1, const float* __restrict__ ow1,
    float* __restrict__ ws) {
  __shared__ __align__(16) float st[128];
  if (threadIdx.x < 128) st[threadIdx.x] = 0.f;
  __syncthreads();
  int lane = threadIdx.x & 31, wave = threadIdx.x >> 5;
  int r0 = (blockIdx.x * 8 + wave) * 16;
  v8f o1[2], a1[2];
  layer1(u, pw1, ws + OFF_CP1, r0, lane, o1);
  layer1(u, ow1, ws + OFF_CO1, r0, lane, a1);
  stats_acc(st, lane, o1);
  stats_acc(st + 64, lane, a1);
  __syncthreads();
  if (threadIdx.x < 128) {
    int t = threadIdx.x;
    int dst = (t < 64) ? (OFF_SO1 + t) : (OFF_SA1 + (t - 64));
    atomicAdd(&ws[dst], st[t]);
  }
}

__global__ void __launch_bounds__(256) k_pass4(const float* __restrict__ u,
    const float* __restrict__ pw1, const float* __restrict__ ow1,
    const float* __restrict__ pw2, const float* __restrict__ ow2,
    float* __restrict__ ws) {
  __shared__ __align__(16) float wP[1024], wO[1024];
  __shared__ __align__(16) float tO[8][16 * LDSW], tA[8][16 * LDSW];
  __shared__ __align__(16) float st[128];
  for (int i = threadIdx.x; i < 1024; i += 256) { wP[i] = pw2[i]; wO[i] = ow2[i]; }
  if (threadIdx.x < 128) st[threadIdx.x] = 0.f;
  __syncthreads();
  int lane = threadIdx.x & 31, wave = threadIdx.x >> 5;
  int r0 = (blockIdx.x * 8 + wave) * 16;
  v8f o1[2], a1[2];
  layer1(u, pw1, ws + OFF_CP1, r0, lane, o1);
  layer1(u, ow1, ws + OFF_CO1, r0, lane, a1);
  store_tile(tO[wave], lane, 0, o1[0]); store_tile(tO[wave], lane, 16, o1[1]);
  store_tile(tA[wave], lane, 0, a1[0]); store_tile(tA[wave], lane, 16, a1[1]);
  __syncthreads();
  v8f o2[2], a2[2];
  gemm32(tO[wave], wP, ws + OFF_CP2, lane, o2);
  gemm32(tA[wave], wO, ws + OFF_CO2, lane, a2);
  v8f rf[2];
  #pragma unroll
  for (int t = 0; t < 2; t++)
    #pragma unroll
    for (int i = 0; i < 8; i++) {
      rf[t][i] = o1[t][i] + fast_tanh(o2[t][i]);
      a2[t][i] = fast_tanh(a2[t][i]);
    }
  stats_acc(st, lane, rf);
  stats_acc(st + 64, lane, a2);
  __syncthreads();
  if (threadIdx.x < 128) {
    int t = threadIdx.x;
    int dst = (t < 64) ? (OFF_SR + t) : (OFF_SA2 + (t - 64));
    atomicAdd(&ws[dst], st[t]);
  }
}

__global__ void __launch_bounds__(256) k_pass6(const float* __restrict__ u,
    const int* __restrict__ mask,
    const float* __restrict__ pw1, const float* __restrict__ ow1,
    const float* __restrict__ pw2, const float* __restrict__ ow2,
    const float* __restrict__ pw3, const float* __restrict__ ow3,
    float* __restrict__ ws) {
  __shared__ __align__(16) float wP2[1024], wO2[1024], wP3[1024], wO3[1024];
  __shared__ __align__(16) float bufA[8][16 * LDSW], bufB[8][16 * LDSW],
                                 bufC[8][16 * LDSW];
  __shared__ float rowmax[8][16], rowinv[8][16];
  __shared__ int mflag[8][16];
  __shared__ float pool[32];
  for (int i = threadIdx.x; i < 1024; i += 256) {
    wP2[i] = pw2[i]; wO2[i] = ow2[i]; wP3[i] = pw3[i]; wO3[i] = ow3[i];
  }
  if (threadIdx.x < 32) pool[threadIdx.x] = 0.f;
  __syncthreads();
  int lane = threadIdx.x & 31, wave = threadIdx.x >> 5;
  int b = blockIdx.x;
  int r0 = b * 128 + wave * 16;

  // ---- layer 1 (both paths), K=4 WMMA ----
  v8f o1[2], a1[2];
  layer1(u, pw1, ws + OFF_CP1, r0, lane, o1);
  layer1(u, ow1, ws + OFF_CO1, r0, lane, a1);
  store_tile(bufA[wave], lane, 0, o1[0]); store_tile(bufA[wave], lane, 16, o1[1]);
  store_tile(bufB[wave], lane, 0, a1[0]); store_tile(bufB[wave], lane, 16, a1[1]);
  if (lane < 16) mflag[wave][lane] = mask[r0 + lane];
  __syncthreads();

  // ---- layer 2 ----
  v8f o2[2], a2[2];
  gemm32(bufA[wave], wP2, ws + OFF_CP2, lane, o2);
  gemm32(bufB[wave], wO2, ws + OFF_CO2, lane, a2);
  v8f rf[2];
  #pragma unroll
  for (int t = 0; t < 2; t++)
    #pragma unroll
    for (int i = 0; i < 8; i++) {
      rf[t][i] = o1[t][i] + fast_tanh(o2[t][i]);
      a2[t][i] = fast_tanh(a2[t][i]);
    }
  __syncthreads();
  store_tile(bufA[wave], lane, 0, rf[0]); store_tile(bufA[wave], lane, 16, rf[1]);
  store_tile(bufB[wave], lane, 0, a2[0]); store_tile(bufB[wave], lane, 16, a2[1]);
  __syncthreads();

  // ---- layer 3 ----
  v8f o3[2], a3[2];
  gemm32(bufA[wave], wP3, ws + OFF_CP3, lane, o3);
  gemm32(bufB[wave], wO3, ws + OFF_CO3, lane, a3);   // no tanh (omega out)
  v8f mf[2];
  #pragma unroll
  for (int t = 0; t < 2; t++)
    #pragma unroll
    for (int i = 0; i < 8; i++) mf[t][i] = rf[t][i] + fast_tanh(o3[t][i]);
  store_tile(bufC[wave], lane, 0, a3[0]); store_tile(bufC[wave], lane, 16, a3[1]);
  __syncthreads();

  // ---- per-row softmax statistics over 32 cols ----
  if (lane < 16) {
    const float* row = &bufC[wave][lane * LDSW];
    float mx = row[0];
    for (int c = 1; c < 32; c++) mx = fmaxf(mx, row[c]);
    float sm = 0.f;
    for (int c = 0; c < 32; c++) sm += __expf(row[c] - mx);
    rowmax[wave][lane] = mx;
    rowinv[wave][lane] = 1.0f / sm;
  }
  __syncthreads();

  // ---- masked attention pooling: lane owns fixed column ----
  #pragma unroll
  for (int t = 0; t < 2; t++) {
    int col = t * 16 + (lane & 15);
    int rb = (lane >> 4) * 8;
    float acc = 0.f;
    #pragma unroll
    for (int j = 0; j < 8; j++) {
      int row = rb + j;
      float attn = (mflag[wave][row] != 0)
          ? __expf(a3[t][j] - rowmax[wave][row]) * rowinv[wave][row]
          : -__builtin_inff();
      acc += mf[t][j] * attn;
    }
    atomicAdd(&pool[col], acc);
  }
  __syncthreads();
  if (threadIdx.x < 32) {
    float v = pool[threadIdx.x];
    ws[OFF_X + b * 32 + threadIdx.x] = v;
    atomicAdd(&ws[OFF_SX + threadIdx.x], v);
    atomicAdd(&ws[OFF_SX + 32 + threadIdx.x], v * v);
  }
}

// Theta-head BN + GEMM(32x32) + tanh + stats; input rows are global (stride 32)
__global__ void __launch_bounds__(256) k_tgemm(const float* __restrict__ in,
    const float* __restrict__ w, float* __restrict__ ws,
    int coefOff, int outOff, int statsOff) {
  __shared__ __align__(16) float st[64];
  if (threadIdx.x < 64) st[threadIdx.x] = 0.f;
  __syncthreads();
  int lane = threadIdx.x & 31, wave = threadIdx.x >> 5;
  int r0 = (blockIdx.x * 8 + wave) * 16;
  const float* coef = ws + coefOff;
  int r = lane & 15;
  int koff = (lane >> 4) * 2;
  v8f d[2]; d[0] = v8zero(); d[1] = v8zero();
  #pragma unroll
  for (int k = 0; k < 8; k++) {
    int cc = 4 * k + koff;
    v2f raw = *(const v2f*)(in + (r0 + r) * 32 + cc);
    v2f a = bn2v(raw, coef, 32, cc);
    v2f b0 = *(const v2f*)(w + r * 32 + cc);
    v2f b1 = *(const v2f*)(w + (16 + r) * 32 + cc);
    d[0] = wmma4(a, b0, d[0]);
    d[1] = wmma4(a, b1, d[1]);
  }
  #pragma unroll
  for (int t = 0; t < 2; t++)
    #pragma unroll
    for (int i = 0; i < 8; i++) d[t][i] = fast_tanh(d[t][i]);
  float* out = ws + outOff;
  #pragma unroll
  for (int t = 0; t < 2; t++) {
    int col = t * 16 + r;
    int rb = (lane >> 4) * 8;
    #pragma unroll
    for (int j = 0; j < 8; j++) out[(r0 + rb + j) * 32 + col] = d[t][j];
  }
  stats_acc(st, lane, d);
  __syncthreads();
  if (threadIdx.x < 64) atomicAdd(&ws[statsOff + threadIdx.x], st[threadIdx.x]);
}

__global__ void k_final(const float* __restrict__ ws,
                        const float* __restrict__ w4,
                        const float* __restrict__ b4,
                        float* __restrict__ out) {
  int rIdx = blockIdx.x * blockDim.x + threadIdx.x;
  if (rIdx >= 4096) return;
  const float* coef = ws + OFF_CT4;
  const float* xr = ws + OFF_T3 + rIdx * 32;
  float acc = 0.f;
  #pragma unroll
  for (int c = 0; c < 32; c++) acc += (xr[c] * coef[c] + coef[32 + c]) * w4[c];
  out[rIdx] = acc + b4[0];
}

// ---------------- host ----------------
extern "C" void kernel_launch(void* const* d_in, const int* in_sizes, int n_in,
                              void* d_out, int out_size, void* d_ws, size_t ws_size,
                              hipStream_t stream) {
  (void)in_sizes; (void)n_in; (void)out_size; (void)ws_size;
  const float* u       = (const float*)d_in[0];
  const int*   mask    = (const int*)d_in[1];
  const float* p_bn1_g = (const float*)d_in[2];
  const float* p_bn1_b = (const float*)d_in[3];
  const float* p_w1    = (const float*)d_in[4];
  const float* p_bn2_g = (const float*)d_in[5];
  const float* p_bn2_b = (const float*)d_in[6];
  const float* p_w2    = (const float*)d_in[7];
  const float* p_bn3_g = (const float*)d_in[8];
  const float* p_bn3_b = (const float*)d_in[9];
  const float* p_w3    = (const float*)d_in[10];
  const float* o_bn1_g = (const float*)d_in[11];
  const float* o_bn1_b = (const float*)d_in[12];
  const float* o_w1    = (const float*)d_in[13];
  const float* o_bn2_g = (const float*)d_in[14];
  const float* o_bn2_b = (const float*)d_in[15];
  const float* o_w2    = (const float*)d_in[16];
  const float* o_bn3_g = (const float*)d_in[17];
  const float* o_bn3_b = (const float*)d_in[18];
  const float* o_w3    = (const float*)d_in[19];
  const float* t_bn1_g = (const float*)d_in[20];
  const float* t_bn1_b = (const float*)d_in[21];
  const float* t_w1    = (const float*)d_in[22];
  const float* t_bn2_g = (const float*)d_in[23];
  const float* t_bn2_b = (const float*)d_in[24];
  const float* t_w2    = (const float*)d_in[25];
  const float* t_bn3_g = (const float*)d_in[26];
  const float* t_bn3_b = (const float*)d_in[27];
  const float* t_w3    = (const float*)d_in[28];
  const float* t_bn4_g = (const float*)d_in[29];
  const float* t_bn4_b = (const float*)d_in[30];
  const float* t_w4    = (const float*)d_in[31];
  const float* t_b4    = (const float*)d_in[32];

  float* ws = (float*)d_ws;
  const int Bt = 4096, N = 128;
  const int M = Bt * N;                       // 524288
  const float invM = 1.0f / (float)M;
  const float invB = 1.0f / (float)Bt;

  k_zero<<<1, 256, 0, stream>>>(ws);
  k_stats_u<<<256, 256, 0, stream>>>(u, ws, M * 4);
  k_coef<<<1, 32, 0, stream>>>(ws, OFF_SU, p_bn1_g, p_bn1_b, OFF_CP1, 4, invM);
  k_coef<<<1, 32, 0, stream>>>(ws, OFF_SU, o_bn1_g, o_bn1_b, OFF_CO1, 4, invM);

  k_pass2<<<M / 128, 256, 0, stream>>>(u, p_w1, o_w1, ws);
  k_coef<<<1, 32, 0, stream>>>(ws, OFF_SO1, p_bn2_g, p_bn2_b, OFF_CP2, 32, invM);
  k_coef<<<1, 32, 0, stream>>>(ws, OFF_SA1, o_bn2_g, o_bn2_b, OFF_CO2, 32, invM);

  k_pass4<<<M / 128, 256, 0, stream>>>(u, p_w1, o_w1, p_w2, o_w2, ws);
  k_coef<<<1, 32, 0, stream>>>(ws, OFF_SR,  p_bn3_g, p_bn3_b, OFF_CP3, 32, invM);
  k_coef<<<1, 32, 0, stream>>>(ws, OFF_SA2, o_bn3_g, o_bn3_b, OFF_CO3, 32, invM);

  k_pass6<<<Bt, 256, 0, stream>>>(u, mask, p_w1, o_w1, p_w2, o_w2, p_w3, o_w3, ws);

  k_coef<<<1, 32, 0, stream>>>(ws, OFF_SX, t_bn1_g, t_bn1_b, OFF_CT1, 32, invB);
  k_tgemm<<<Bt / 128, 256, 0, stream>>>(ws + OFF_X, t_w1, ws, OFF_CT1, OFF_T1, OFF_ST1);
  k_coef<<<1, 32, 0, stream>>>(ws, OFF_ST1, t_bn2_g, t_bn2_b, OFF_CT2, 32, invB);
  k_tgemm<<<Bt / 128, 256, 0, stream>>>(ws + OFF_T1, t_w2, ws, OFF_CT2, OFF_T2, OFF_ST2);
  k_coef<<<1, 32, 0, stream>>>(ws, OFF_ST2, t_bn3_g, t_bn3_b, OFF_CT3, 32, invB);
  k_tgemm<<<Bt / 128, 256, 0, stream>>>(ws + OFF_T2, t_w3, ws, OFF_CT3, OFF_T3, OFF_ST3);
  k_coef<<<1, 32, 0, stream>>>(ws, OFF_ST3, t_bn4_g, t_bn4_b, OFF_CT4, 32, invB);

  k_final<<<16, 256, 0, stream>>>(ws, t_w4, t_b4, (float*)d_out);
}